// LinearAttention_81458349736744
// MI455X (gfx1250) — compile-verified
//
#include <hip/hip_runtime.h>

typedef float v2f __attribute__((ext_vector_type(2)));
typedef float v8f __attribute__((ext_vector_type(8)));

#define NTOK 16384
#define NCHUNK 8
#define CHUNK_TOK 2048
#define TILE_N 64
#define SCALE_Q 0.17677669529663687f /* 32^-0.5 */

// ---------------- WMMA f32 16x16x4 fragment helpers ----------------
// Load A fragment (row-major [M=16][K]) into registers: K/4 x v2f
template <int K>
__device__ __forceinline__ void load_a_rm(v2f* af, const float* A, int ldA,
                                          int lane) {
  const int row = lane & 15;
  const int g2  = (lane >> 4) << 1;
#pragma unroll
  for (int i = 0; i < K / 4; ++i) {
    af[i].x = A[row * ldA + i * 4 + g2];
    af[i].y = A[row * ldA + i * 4 + g2 + 1];
  }
}

// Load A fragment from transposed storage [K][M=16]
template <int K>
__device__ __forceinline__ void load_a_tr(v2f* af, const float* At, int ldAt,
                                          int lane) {
  const int row = lane & 15;
  const int g2  = (lane >> 4) << 1;
#pragma unroll
  for (int i = 0; i < K / 4; ++i) {
    af[i].x = At[(i * 4 + g2) * ldAt + row];
    af[i].y = At[(i * 4 + g2 + 1) * ldAt + row];
  }
}

// D += Afrag x B, B row-major [K][N=16].
// B fragments preloaded in blocks of 16 k-steps (distinct regs) so the
// scheduler can overlap DS returns with the WMMA burst (partial dscnt waits).
template <int K>
__device__ __forceinline__ v8f wmma_fb(const v2f* af, const float* B, int ldB,
                                       v8f acc, int lane) {
  const int row = lane & 15;
  const int g2  = (lane >> 4) << 1;
  constexpr int NB = K / 4;
#pragma unroll
  for (int h = 0; h < NB; h += 16) {
    constexpr int BLK = NB < 16 ? NB : 16;
    v2f bf[BLK];
#pragma unroll
    for (int i = 0; i < BLK; ++i) {
      int kk = (h + i) * 4 + g2;
      bf[i].x = B[kk * ldB + row];
      bf[i].y = B[(kk + 1) * ldB + row];
    }
#pragma unroll
    for (int i = 0; i < BLK; ++i)
      acc = __builtin_amdgcn_wmma_f32_16x16x4_f32(false, af[h + i], false,
                                                  bf[i], (short)0, acc, false,
                                                  false);
  }
  return acc;
}

// D += Afrag x B^T, B supplied transposed as [N=16][K]
template <int K>
__device__ __forceinline__ v8f wmma_fbT(const v2f* af, const float* Bt,
                                        int ldBt, v8f acc, int lane) {
  const int row = lane & 15;
  const int g2  = (lane >> 4) << 1;
  constexpr int NB = K / 4;
#pragma unroll
  for (int h = 0; h < NB; h += 16) {
    constexpr int BLK = NB < 16 ? NB : 16;
    v2f bf[BLK];
#pragma unroll
    for (int i = 0; i < BLK; ++i) {
      int kk = (h + i) * 4 + g2;
      bf[i].x = Bt[row * ldBt + kk];
      bf[i].y = Bt[row * ldBt + kk + 1];
    }
#pragma unroll
    for (int i = 0; i < BLK; ++i)
      acc = __builtin_amdgcn_wmma_f32_16x16x4_f32(false, af[h + i], false,
                                                  bf[i], (short)0, acc, false,
                                                  false);
  }
  return acc;
}

__device__ __forceinline__ void store_tile(float* D, int ldD, v8f acc, int lane) {
  const int row = lane & 15;
  const int g8  = (lane >> 4) << 3;
#pragma unroll
  for (int r = 0; r < 8; ++r) D[(r + g8) * ldD + row] = acc[r];
}

__device__ __forceinline__ void ld_tile(float* dst, int ldd,
                                        const float* src, int lds_,
                                        int rows, int cols, int tid, int nthr) {
  for (int idx = tid; idx < rows * cols; idx += nthr) {
    int r = idx / cols, cc = idx - r * cols;
    dst[r * ldd + cc] = src[(size_t)r * lds_ + cc];
  }
}

// ---------------- K1: k row stats (per-chunk partial max / sumexp) ----------------
// spart layout: [b][chunk][row(128)][2]
__global__ void LA_k1_kstats(const float* __restrict__ x,
                             const float* __restrict__ Wqkv,
                             float* __restrict__ spart) {
  extern __shared__ float sm[];
  float* W  = sm;              // 128 x 129
  float* XT = W + 128 * 129;   // 128 x 65
  float* KT = XT + 128 * 65;   // 128 x 65
  const int tid = threadIdx.x, lane = tid & 31, wid = tid >> 5;
  const int chunk = blockIdx.x, b = blockIdx.y;
  const v8f Z = {0, 0, 0, 0, 0, 0, 0, 0};

  ld_tile(W, 129, Wqkv + 128 * 128, 128, 128, 128, tid, 256); // Wk

  float rmax = -3.0e38f, rsum = 0.0f;
  for (int st = 0; st < CHUNK_TOK / TILE_N; ++st) {
    const int n0 = chunk * CHUNK_TOK + st * TILE_N;
    __syncthreads();
    ld_tile(XT, 65, x + (size_t)b * 128 * NTOK + n0, NTOK, 128, 64, tid, 256);
    __syncthreads();
    {
      v2f af[32];                                     // A = Wk row-block (wid)
      load_a_rm<128>(af, W + wid * 16 * 129, 129, lane);
#pragma unroll
      for (int nt = 0; nt < 4; ++nt) {
        v8f acc = wmma_fb<128>(af, XT + nt * 16, 65, Z, lane);
        store_tile(KT + wid * 16 * 65 + nt * 16, 65, acc, lane);
      }
    }
    __syncthreads();
    if (tid < 128) {
      float tmax = -3.0e38f;
      for (int j = 0; j < 64; ++j) tmax = fmaxf(tmax, KT[tid * 65 + j]);
      float nm = fmaxf(rmax, tmax);
      rsum *= __expf(rmax - nm);
      for (int j = 0; j < 64; ++j) rsum += __expf(KT[tid * 65 + j] - nm);
      rmax = nm;
    }
  }
  if (tid < 128) {
    size_t o = (((size_t)b * NCHUNK + chunk) * 128 + tid) * 2;
    spart[o] = rmax;
    spart[o + 1] = rsum;
  }
}

// ---------------- K2: combine stats -> (max, 1/sum) ----------------
__global__ void LA_k2_combine(const float* __restrict__ spart,
                              float* __restrict__ sfin) {
  int idx = blockIdx.x * 256 + threadIdx.x;
  if (idx >= 32 * 128) return;
  int b = idx >> 7, r = idx & 127;
  float M = -3.0e38f;
  for (int ch = 0; ch < NCHUNK; ++ch)
    M = fmaxf(M, spart[(((size_t)b * NCHUNK + ch) * 128 + r) * 2]);
  float S = 0.0f;
  for (int ch = 0; ch < NCHUNK; ++ch) {
    size_t o = (((size_t)b * NCHUNK + ch) * 128 + r) * 2;
    S += spart[o + 1] * __expf(spart[o] - M);
  }
  sfin[((size_t)b * 128 + r) * 2] = M;
  sfin[((size_t)b * 128 + r) * 2 + 1] = 1.0f / S;
}

// ---------------- K3: ctx partials per chunk ----------------
// ctxp layout: [chunk][b][h][d(32)][e(32)]
__global__ void LA_k3_ctx(const float* __restrict__ x,
                          const float* __restrict__ Wqkv,
                          const float* __restrict__ sfin,
                          float* __restrict__ ctxp) {
  extern __shared__ float sm[];
  float* W   = sm;                // 256 x 129 (Wk rows 0-127, Wv rows 128-255)
  float* XT  = W + 256 * 129;     // 128 x 65
  float* KV  = XT + 128 * 65;     // 256 x 65 (k_sm rows 0-127, v' rows 128-255)
  float* RMX = KV + 256 * 65;     // 128
  float* RIV = RMX + 128;         // 128
  const int tid = threadIdx.x, lane = tid & 31, wid = tid >> 5;
  const int chunk = blockIdx.x, b = blockIdx.y;
  const v8f Z = {0, 0, 0, 0, 0, 0, 0, 0};
  const float invn = 1.0f / (float)NTOK;

  ld_tile(W, 129, Wqkv + 128 * 128, 128, 256, 128, tid, 256); // Wk|Wv
  if (tid < 128) {
    RMX[tid] = sfin[((size_t)b * 128 + tid) * 2];
    RIV[tid] = sfin[((size_t)b * 128 + tid) * 2 + 1];
  }

  // ctx tile ownership: wave -> fixed (h, dt); two accumulators for et=0/1
  const int h_c  = wid >> 1;
  const int dt_c = wid & 1;
  v8f acc0 = Z, acc1 = Z;

  for (int st = 0; st < CHUNK_TOK / TILE_N; ++st) {
    const int n0 = chunk * CHUNK_TOK + st * TILE_N;
    __syncthreads();
    ld_tile(XT, 65, x + (size_t)b * 128 * NTOK + n0, NTOK, 128, 64, tid, 256);
    __syncthreads();
    // kv tile: wave handles M-tiles {wid, wid+8}, A cached across 4 N-tiles
#pragma unroll
    for (int i = 0; i < 2; ++i) {
      int mt = wid + i * 8;
      v2f af[32];
      load_a_rm<128>(af, W + mt * 16 * 129, 129, lane);
#pragma unroll
      for (int nt = 0; nt < 4; ++nt) {
        v8f a = wmma_fb<128>(af, XT + nt * 16, 65, Z, lane);
        store_tile(KV + mt * 16 * 65 + nt * 16, 65, a, lane);
      }
    }
    __syncthreads();
    for (int idx = tid; idx < 128 * 64; idx += 256) {
      int r = idx >> 6, j = idx & 63;
      KV[r * 65 + j] = __expf(KV[r * 65 + j] - RMX[r]) * RIV[r];
      KV[(128 + r) * 65 + j] *= invn;
    }
    __syncthreads();
    {
      v2f af[16];                                    // k_sm rows (h_c, dt_c)
      load_a_rm<64>(af, KV + (h_c * 32 + dt_c * 16) * 65, 65, lane);
      acc0 = wmma_fbT<64>(af, KV + (128 + h_c * 32) * 65, 65, acc0, lane);
      acc1 = wmma_fbT<64>(af, KV + (128 + h_c * 32 + 16) * 65, 65, acc1, lane);
    }
  }
  {
    float* base0 = ctxp + (((size_t)chunk * 32 + b) * 4 + h_c) * 1024 +
                   dt_c * 16 * 32;
    store_tile(base0, 32, acc0, lane);
    store_tile(base0 + 16, 32, acc1, lane);
  }
}

// ---------------- K4: reduce ctx partials (fixed order) ----------------
__global__ void LA_k4_ctxreduce(const float* __restrict__ ctxp,
                                float* __restrict__ ctxf) {
  int idx = blockIdx.x * 256 + threadIdx.x; // 131072 total
  float s = 0.0f;
  for (int ch = 0; ch < NCHUNK; ++ch) s += ctxp[(size_t)ch * 131072 + idx];
  ctxf[idx] = s;
}

// ---------------- K5: q softmax, out = ctx^T q, y = Wout out + b, LayerNorm ----------------
__global__ void LA_k5_out(const float* __restrict__ x,
                          const float* __restrict__ Wqkv,
                          const float* __restrict__ Wout,
                          const float* __restrict__ bout,
                          const float* __restrict__ g,
                          const float* __restrict__ ctxf,
                          float* __restrict__ out) {
  extern __shared__ float sm[];
  float* W  = sm;              // 128 x 129 : Wq then Wout
  float* XT = W + 128 * 129;   // 128 x 65  : x tile, later y tile
  float* QT = XT + 128 * 65;   // 128 x 65  : q tile
  float* OT = QT + 128 * 65;   // 128 x 65  : out tile
  float* CT = OT + 128 * 65;   // 4 x 32 x 33 = 4224
  float* GL = CT + 4224;       // 128
  float* BL = GL + 128;        // 128
  float* MU = BL + 128;        // 64
  float* RS = MU + 64;         // 64
  const int tid = threadIdx.x, lane = tid & 31, wid = tid >> 5;
  const int ntile = blockIdx.x, b = blockIdx.y;
  const int n0 = ntile * TILE_N;
  const v8f Z = {0, 0, 0, 0, 0, 0, 0, 0};

  ld_tile(W, 129, Wqkv, 128, 128, 128, tid, 256); // Wq
  ld_tile(XT, 65, x + (size_t)b * 128 * NTOK + n0, NTOK, 128, 64, tid, 256);
  for (int idx = tid; idx < 4096; idx += 256) {
    int h = idx >> 10, de = idx & 1023;
    int d = de >> 5, e = de & 31;
    CT[h * 1056 + d * 33 + e] = ctxf[(size_t)b * 4096 + idx];
  }
  if (tid < 128) { GL[tid] = g[tid]; BL[tid] = bout[tid]; }
  __syncthreads();

  // q raw = Wq * x (A cached per wave across 4 N-tiles)
  {
    v2f af[32];
    load_a_rm<128>(af, W + wid * 16 * 129, 129, lane);
#pragma unroll
    for (int nt = 0; nt < 4; ++nt) {
      v8f acc = wmma_fb<128>(af, XT + nt * 16, 65, Z, lane);
      store_tile(QT + wid * 16 * 65 + nt * 16, 65, acc, lane);
    }
  }
  __syncthreads();

  // q softmax over the 32 head-channels per (head, column), times scale
  {
    int col = tid & 63, h = tid >> 6;
    int base = (h * 32) * 65 + col;
    float m = -3.0e38f;
    for (int d = 0; d < 32; ++d) m = fmaxf(m, QT[base + d * 65]);
    float s = 0.0f;
    for (int d = 0; d < 32; ++d) s += __expf(QT[base + d * 65] - m);
    float sc = SCALE_Q / s;
    for (int d = 0; d < 32; ++d) QT[base + d * 65] = __expf(QT[base + d * 65] - m) * sc;
  }
  __syncthreads();

  ld_tile(W, 129, Wout, 128, 128, 128, tid, 256); // Wq dead; overlap with out GEMM

  // out[e,n] = sum_d ctx[d,e] * q_sm[d,n]; wave owns (h, et), ctx^T frag cached
  {
    const int h  = wid >> 1;
    const int et = wid & 1;
    v2f af[8];
    load_a_tr<32>(af, CT + h * 1056 + et * 16, 33, lane);
#pragma unroll
    for (int nt = 0; nt < 4; ++nt) {
      v8f acc = wmma_fb<32>(af, QT + (h * 32) * 65 + nt * 16, 65, Z, lane);
      store_tile(OT + (h * 32 + et * 16) * 65 + nt * 16, 65, acc, lane);
    }
  }
  __syncthreads();

  // y = Wout * out + bout -> XT
  {
    const int row = lane & 15, g8 = (lane >> 4) << 3;
    v2f af[32];
    load_a_rm<128>(af, W + wid * 16 * 129, 129, lane);
#pragma unroll
    for (int nt = 0; nt < 4; ++nt) {
      v8f acc = wmma_fb<128>(af, OT + nt * 16, 65, Z, lane);
#pragma unroll
      for (int r = 0; r < 8; ++r) {
        int m = wid * 16 + r + g8;
        XT[m * 65 + nt * 16 + row] = acc[r] + BL[m];
      }
    }
  }
  __syncthreads();

  // LayerNorm over 128 channels per column
  if (tid < 64) {
    float s = 0.0f, s2 = 0.0f;
    for (int ch = 0; ch < 128; ++ch) {
      float v = XT[ch * 65 + tid];
      s += v; s2 += v * v;
    }
    float mu = s * (1.0f / 128.0f);
    float var = s2 * (1.0f / 128.0f) - mu * mu;
    MU[tid] = mu;
    RS[tid] = rsqrtf(var + 1e-5f);
  }
  __syncthreads();
  for (int idx = tid; idx < 8192; idx += 256) {
    int ch = idx >> 6, j = idx & 63;
    out[((size_t)b * 128 + ch) * NTOK + n0 + j] =
        (XT[ch * 65 + j] - MU[j]) * RS[j] * GL[ch];
  }
}

// ---------------- host ----------------
extern "C" void kernel_launch(void* const* d_in, const int* in_sizes, int n_in,
                              void* d_out, int out_size, void* d_ws, size_t ws_size,
                              hipStream_t stream) {
  const float* x    = (const float*)d_in[0];
  const float* Wqkv = (const float*)d_in[1];
  const float* Wout = (const float*)d_in[2];
  const float* bout = (const float*)d_in[3];
  const float* g    = (const float*)d_in[4];
  float* out = (float*)d_out;

  float* ws    = (float*)d_ws;
  float* spart = ws;                          // 32*8*128*2   = 65536
  float* sfin  = spart + 32 * 8 * 128 * 2;    // 32*128*2     = 8192
  float* ctxp  = sfin + 32 * 128 * 2;         // 8*32*4*32*32 = 1048576
  float* ctxf  = ctxp + 8 * 32 * 4 * 32 * 32; // 131072

  const int SM1 = (128 * 129 + 2 * 128 * 65) * 4;
  const int SM3 = (256 * 129 + 128 * 65 + 256 * 65 + 256) * 4;
  const int SM5 = (128 * 129 + 3 * 128 * 65 + 4224 + 384) * 4;

  (void)hipFuncSetAttribute((const void*)LA_k1_kstats,
                            hipFuncAttributeMaxDynamicSharedMemorySize, SM1);
  (void)hipFuncSetAttribute((const void*)LA_k3_ctx,
                            hipFuncAttributeMaxDynamicSharedMemorySize, SM3);
  (void)hipFuncSetAttribute((const void*)LA_k5_out,
                            hipFuncAttributeMaxDynamicSharedMemorySize, SM5);

  LA_k1_kstats<<<dim3(NCHUNK, 32), 256, SM1, stream>>>(x, Wqkv, spart);
  LA_k2_combine<<<dim3(16), 256, 0, stream>>>(spart, sfin);
  LA_k3_ctx<<<dim3(NCHUNK, 32), 256, SM3, stream>>>(x, Wqkv, sfin, ctxp);
  LA_k4_ctxreduce<<<dim3(512), 256, 0, stream>>>(ctxp, ctxf);
  LA_k5_out<<<dim3(NTOK / TILE_N, 32), 256, SM5, stream>>>(x, Wqkv, Wout, bout,
                                                           g, ctxf, out);
}